// GNN_52939766890541
// MI455X (gfx1250) — compile-verified
//
#include <hip/hip_runtime.h>
#include <math.h>

typedef __attribute__((ext_vector_type(2))) float v2f;
typedef __attribute__((ext_vector_type(8))) float v8f;

#define ALPHA 0.5f
#define SCAN_BS 1024

// ---------------- zero fill (int4 grid-stride) ----------------
__global__ void zero4_kernel(int4* __restrict__ p, long n4) {
    long i = (long)blockIdx.x * blockDim.x + threadIdx.x;
    long stride = (long)gridDim.x * blockDim.x;
    int4 z; z.x = 0; z.y = 0; z.z = 0; z.w = 0;
    for (; i < n4; i += stride) p[i] = z;
}

// ---------------- integer degree accumulation ----------------
__global__ void degree_kernel(const int* __restrict__ row, const int* __restrict__ col,
                              int* __restrict__ degS, int* __restrict__ degD, int E) {
    int e = blockIdx.x * blockDim.x + threadIdx.x;
    if (e >= E) return;
    atomicAdd(&degS[row[e]], 1);
    atomicAdd(&degD[col[e]], 1);
}

// ---------------- exclusive scan: stage 1 (per-block inclusive) ----------------
__global__ void scan_block_kernel(const int* __restrict__ in, int* __restrict__ partial,
                                  int* __restrict__ bsums, int n) {
    __shared__ int sh[SCAN_BS];
    int t = threadIdx.x;
    int i = blockIdx.x * SCAN_BS + t;
    sh[t] = (i < n) ? in[i] : 0;
    __syncthreads();
    for (int off = 1; off < SCAN_BS; off <<= 1) {
        int v = (t >= off) ? sh[t - off] : 0;
        __syncthreads();
        sh[t] += v;
        __syncthreads();
    }
    if (i < n) partial[i] = sh[t];
    if (t == SCAN_BS - 1) bsums[blockIdx.x] = sh[t];
}

// ---------------- exclusive scan: stage 2 (block sums, single block) --------
__global__ void scan_top_kernel(int* __restrict__ bsums, int nb) {
    __shared__ int sh[SCAN_BS];
    int t = threadIdx.x;
    sh[t] = (t < nb) ? bsums[t] : 0;
    __syncthreads();
    for (int off = 1; off < SCAN_BS; off <<= 1) {
        int v = (t >= off) ? sh[t - off] : 0;
        __syncthreads();
        sh[t] += v;
        __syncthreads();
    }
    if (t < nb) bsums[t] = (t == 0) ? 0 : sh[t - 1];   // exclusive
}

// ---------------- exclusive scan: stage 3 (offsets + cursor init) -----------
__global__ void finalize_offsets_kernel(const int* __restrict__ partial,
                                        const int* __restrict__ bsumsExcl,
                                        int* __restrict__ offs, int* __restrict__ cur, int n) {
    int i = blockIdx.x * blockDim.x + threadIdx.x;
    if (i < n) {
        int incl = partial[i] + bsumsExcl[i >> 10];     // global inclusive at i
        offs[i + 1] = incl;
        int excl = incl - (partial[i] - ((i & (SCAN_BS - 1)) ? 0 : 0));
        (void)excl;
    }
    if (i == 0) offs[0] = 0;
}

__global__ void copy_int_kernel(const int* __restrict__ a, int* __restrict__ b, int n) {
    int i = blockIdx.x * blockDim.x + threadIdx.x;
    if (i < n) b[i] = a[i];
}

// ---------------- CSR bucket fill (weight computed inline) ----------------
__global__ void csr_fill_kernel(const int* __restrict__ row, const int* __restrict__ col,
                                const int* __restrict__ degS, const int* __restrict__ degD,
                                int* __restrict__ curS, int* __restrict__ curD,
                                int2* __restrict__ ecsrS, int2* __restrict__ ecsrD, int E) {
    int e = blockIdx.x * blockDim.x + threadIdx.x;
    if (e >= E) return;
    int r = row[e];
    int c = col[e];
    // deg >= 1 guaranteed at incident nodes; reference guard never fires here
    float wv = rsqrtf((float)max(degS[r], 1)) * rsqrtf((float)max(degD[c], 1));
    int wb = __float_as_int(wv);
    int ps = atomicAdd(&curS[r], 1);
    ecsrS[ps] = make_int2(c, wb);
    int pd = atomicAdd(&curD[c], 1);
    ecsrD[pd] = make_int2(r, wb);
}

// ---------------- gather-form SpMM: y[v] = sum_e w_e * x[nbr_e]  (F = 64) ----
// thread = (node, 4-feature chunk); overwrites y (no zero-init, no atomics)
__global__ void spmm_gather_kernel(const int2* __restrict__ ecsr, const int* __restrict__ offs,
                                   const float* __restrict__ x, float* __restrict__ y, int N) {
    long gid = (long)blockIdx.x * blockDim.x + threadIdx.x;
    int v = (int)(gid >> 4);
    if (v >= N) return;
    int c4 = ((int)gid & 15) << 2;   // feature offset 0..60
    int p = offs[v];
    int pend = offs[v + 1];
    float4 acc; acc.x = 0.f; acc.y = 0.f; acc.z = 0.f; acc.w = 0.f;
    for (; p < pend; ++p) {
        int2 en = ecsr[p];                       // one global_load_b64
        float wv = __int_as_float(en.y);
        const float4 xv = *(const float4*)(x + (long)en.x * 64 + c4);
        acc.x += wv * xv.x;
        acc.y += wv * xv.y;
        acc.z += wv * xv.z;
        acc.w += wv * xv.w;
    }
    *(float4*)(y + (long)v * 64 + c4) = acc;
}

// ---------------- blended/padded weight prep (single block) ----------------
__global__ void prep_weights_kernel(const float* __restrict__ Ws1, const float* __restrict__ bs1,
                                    const float* __restrict__ Wd1, const float* __restrict__ bd1,
                                    const float* __restrict__ Ws2, const float* __restrict__ bs2,
                                    const float* __restrict__ Wd2, const float* __restrict__ bd2,
                                    float* __restrict__ Ba1, float* __restrict__ Bb1, float* __restrict__ bias1,
                                    float* __restrict__ Ba2, float* __restrict__ Bb2, float* __restrict__ bias2) {
    const float a = ALPHA, b = 1.0f - ALPHA;
    int t = threadIdx.x;
    for (int i = t; i < 64 * 64; i += blockDim.x) { Ba1[i] = a * Ws1[i]; Bb1[i] = b * Wd1[i]; }
    for (int i = t; i < 64; i += blockDim.x) bias1[i] = a * bs1[i] + b * bd1[i];
    for (int i = t; i < 64 * 48; i += blockDim.x) {
        int k = i / 48, n = i % 48;
        float va = 0.f, vb = 0.f;
        if (n < 40) { va = a * Ws2[k * 40 + n]; vb = b * Wd2[k * 40 + n]; }
        Ba2[i] = va; Bb2[i] = vb;
    }
    for (int i = t; i < 48; i += blockDim.x) bias2[i] = (i < 40) ? (a * bs2[i] + b * bd2[i]) : 0.f;
}

// ---------------- fused dual WMMA linear: out = As@Ba + Ad@Bb + bias -------
// K = 64 fixed. One wave32 per 16x16 tile; blockDim.x = 32 * (Ncols/16);
// grid.x = Mtiles (N rows multiple of 16: 100000 = 6250*16). EXEC all-1s.
template <bool RELU>
__global__ void gcn_linear_kernel(const float* __restrict__ As, const float* __restrict__ Ad,
                                  const float* __restrict__ Ba, const float* __restrict__ Bb,
                                  const float* __restrict__ bias, float* __restrict__ out,
                                  int Ncols, int outLd, int outCols) {
    const int wave = threadIdx.x >> 5;
    const int lane = threadIdx.x & 31;
    const int mBase = blockIdx.x << 4;
    const int nBase = wave << 4;
    const int m = lane & 15;            // A row / B,C column within tile
    const int ko = (lane >> 4) << 1;    // K sub-offset: 0 (lanes 0-15) or 2 (lanes 16-31)

    const float* arow_s = As + (long)(mBase + m) * 64;
    const float* arow_d = Ad + (long)(mBase + m) * 64;

    v8f acc = {};
    for (int k = 0; k < 64; k += 4) {
        v2f a, b;
        // direction 1: ys @ (alpha*Ws)
        a.x = arow_s[k + ko];
        a.y = arow_s[k + ko + 1];
        b.x = Ba[(k + ko) * Ncols + nBase + m];
        b.y = Ba[(k + ko + 1) * Ncols + nBase + m];
        acc = __builtin_amdgcn_wmma_f32_16x16x4_f32(false, a, false, b,
                                                    (short)0, acc, false, false);
        // direction 2: yd @ ((1-alpha)*Wd)
        a.x = arow_d[k + ko];
        a.y = arow_d[k + ko + 1];
        b.x = Bb[(k + ko) * Ncols + nBase + m];
        b.y = Bb[(k + ko + 1) * Ncols + nBase + m];
        acc = __builtin_amdgcn_wmma_f32_16x16x4_f32(false, a, false, b,
                                                    (short)0, acc, false, false);
    }

    const int n = nBase + m;
    const float bv = bias[n];
    const bool nOK = (n < outCols);
    const int mHalf = (lane >> 4) << 3;  // 0 or 8
#pragma unroll
    for (int r = 0; r < 8; ++r) {
        int mm = mBase + r + mHalf;
        float v = acc[r] + bv;
        if (RELU) v = fmaxf(v, 0.f);
        if (nOK) out[(long)mm * outLd + n] = v;
    }
}

// ---------------- row-wise log_softmax over 40 classes, one wave32 per row ----
__global__ void log_softmax40_kernel(float* __restrict__ out, int nRows) {
    int wave = threadIdx.x >> 5;
    int lane = threadIdx.x & 31;
    int rowi = blockIdx.x * (blockDim.x >> 5) + wave;
    if (rowi >= nRows) return;
    float* p = out + (long)rowi * 40;
    float v0 = p[lane];                                   // cols 0..31
    float v1 = (lane < 8) ? p[lane + 32] : -INFINITY;     // cols 32..39
    float mx = fmaxf(v0, v1);
#pragma unroll
    for (int off = 16; off > 0; off >>= 1) mx = fmaxf(mx, __shfl_xor(mx, off, 32));
    float s = expf(v0 - mx) + ((lane < 8) ? expf(v1 - mx) : 0.f);
#pragma unroll
    for (int off = 16; off > 0; off >>= 1) s += __shfl_xor(s, off, 32);
    float lse = mx + logf(s);
    p[lane] = v0 - lse;
    if (lane < 8) p[lane + 32] = v1 - lse;
}

// =====================================================================
extern "C" void kernel_launch(void* const* d_in, const int* in_sizes, int n_in,
                              void* d_out, int out_size, void* d_ws, size_t ws_size,
                              hipStream_t stream) {
    const float* x   = (const float*)d_in[0];
    const int*   ei  = (const int*)d_in[1];
    const float* Ws1 = (const float*)d_in[2];
    const float* bs1 = (const float*)d_in[3];
    const float* Wd1 = (const float*)d_in[4];
    const float* bd1 = (const float*)d_in[5];
    const float* Ws2 = (const float*)d_in[6];
    const float* bs2 = (const float*)d_in[7];
    const float* Wd2 = (const float*)d_in[8];
    const float* bd2 = (const float*)d_in[9];

    const int N = in_sizes[0] / 64;   // 100000
    const int E = in_sizes[1] / 2;    // 1600000
    const int* row = ei;
    const int* col = ei + E;

    // ---- workspace carve-up (4-byte units)
    char* base = (char*)d_ws;
    long off = 0;
    auto take = [&](long nElems) { void* p = base + off; off += nElems * 4L; return p; };

    int*  degS   = (int*)take(N);          // must be adjacent to degD for one zero pass
    int*  degD   = (int*)take(N);
    int*  offsS  = (int*)take(N + 4);
    int*  offsD  = (int*)take(N + 4);
    int*  curS   = (int*)take(N);
    int*  curD   = (int*)take(N);
    int*  partial= (int*)take(N);
    int*  bsums  = (int*)take(SCAN_BS);
    int2* ecsrS  = (int2*)take(2L * E);
    int2* ecsrD  = (int2*)take(2L * E);
    float* ys    = (float*)take((long)N * 64);
    float* yd    = (float*)take((long)N * 64);
    float* h     = (float*)take((long)N * 64);
    float* Ba1   = (float*)take(64 * 64);
    float* Bb1   = (float*)take(64 * 64);
    float* bias1 = (float*)take(64);
    float* Ba2   = (float*)take(64 * 48);
    float* Bb2   = (float*)take(64 * 48);
    float* bias2 = (float*)take(48);

    float* logits = (float*)d_out;    // N x 40

    const int TB = 256;
    const int nbScan = (N + SCAN_BS - 1) / SCAN_BS;   // 98

    // 1) zero degrees (2N ints, contiguous, N multiple of 4)
    zero4_kernel<<<512, TB, 0, stream>>>((int4*)degS, (long)2 * N / 4);

    // 2) degrees
    degree_kernel<<<(E + TB - 1) / TB, TB, 0, stream>>>(row, col, degS, degD, E);

    // 3) exclusive scans -> offsets (S then D), cursors init
    scan_block_kernel<<<nbScan, SCAN_BS, 0, stream>>>(degS, partial, bsums, N);
    scan_top_kernel<<<1, SCAN_BS, 0, stream>>>(bsums, nbScan);
    finalize_offsets_kernel<<<(N + TB - 1) / TB, TB, 0, stream>>>(partial, bsums, offsS, curS, N);
    copy_int_kernel<<<(N + TB - 1) / TB, TB, 0, stream>>>(offsS, curS, N);

    scan_block_kernel<<<nbScan, SCAN_BS, 0, stream>>>(degD, partial, bsums, N);
    scan_top_kernel<<<1, SCAN_BS, 0, stream>>>(bsums, nbScan);
    finalize_offsets_kernel<<<(N + TB - 1) / TB, TB, 0, stream>>>(partial, bsums, offsD, curD, N);
    copy_int_kernel<<<(N + TB - 1) / TB, TB, 0, stream>>>(offsD, curD, N);

    // 4) CSR fill (both directions, weight inline)
    csr_fill_kernel<<<(E + TB - 1) / TB, TB, 0, stream>>>(row, col, degS, degD,
                                                          curS, curD, ecsrS, ecsrD, E);

    // 5) blended weights / biases
    prep_weights_kernel<<<1, 256, 0, stream>>>(Ws1, bs1, Wd1, bd1, Ws2, bs2, Wd2, bd2,
                                               Ba1, Bb1, bias1, Ba2, Bb2, bias2);

    // 6) layer-1 aggregation (gather form, overwrites ys/yd)
    int spBlocks = (int)(((long)N * 16 + TB - 1) / TB);
    spmm_gather_kernel<<<spBlocks, TB, 0, stream>>>(ecsrS, offsS, x, ys, N);
    spmm_gather_kernel<<<spBlocks, TB, 0, stream>>>(ecsrD, offsD, x, yd, N);

    // 7) layer-1 fused linear + ReLU -> h   (WMMA f32 16x16x4)
    gcn_linear_kernel<true><<<N / 16, 128, 0, stream>>>(ys, yd, Ba1, Bb1, bias1, h,
                                                        64, 64, 64);

    // 8) layer-2 aggregation on h
    spmm_gather_kernel<<<spBlocks, TB, 0, stream>>>(ecsrS, offsS, h, ys, N);
    spmm_gather_kernel<<<spBlocks, TB, 0, stream>>>(ecsrD, offsD, h, yd, N);

    // 9) layer-2 fused linear -> logits (Ncols padded 48, store 40)
    gcn_linear_kernel<false><<<N / 16, 96, 0, stream>>>(ys, yd, Ba2, Bb2, bias2, logits,
                                                        48, 40, 40);

    // 10) log_softmax in place (one wave per row, 8 waves per block)
    log_softmax40_kernel<<<(N + 7) / 8, 256, 0, stream>>>(logits, N);
}